// DKVMN_7576322310262
// MI455X (gfx1250) — compile-verified
//
#include <hip/hip_runtime.h>
#include <hip/hip_bf16.h>

typedef __attribute__((ext_vector_type(2))) float v2f;
typedef __attribute__((ext_vector_type(8))) float v8f;

#define S_DIM 128
#define M_SIZE 64
#define NUM_Q 10000
#define NTOK 32768          // B*T = 64*512

#define LDS_STRIDE1 132     // 128 + 4 pad floats  (132 % 64 banks = 4 -> conflict-free)
#define LDS_STRIDE3 260     // 256 + 4 pad floats  (260 % 64 banks = 4 -> conflict-free)
#define MK_BASE 0
#define EW_BASE (64 * LDS_STRIDE1)
#define AW_BASE (192 * LDS_STRIDE1)

extern __shared__ float smem[];

// D = A(16x4 f32) * B(4x16 f32) + C(16x16 f32), wave32
__device__ __forceinline__ v8f wmma4(v2f a, v2f b, v8f c) {
    return __builtin_amdgcn_wmma_f32_16x16x4_f32(
        false, a, false, b, (short)0, c, false, false);
}

__device__ __forceinline__ float sigmoidf_(float x) {
    return 1.0f / (1.0f + __expf(-x));
}

// Async-stage `rows` rows of `row_len` floats (row_len multiple of 4) from
// global into LDS with padded stride `lds_stride` (floats). 16B granules.
__device__ __forceinline__ void async_stage_rows(
    const float* __restrict__ g, int row_len, int lds_base_floats,
    int lds_stride, int rows, int tid, int nthreads)
{
    const int cpr    = row_len >> 2;       // 16B chunks per row
    const int chunks = rows * cpr;
    for (int c = tid; c < chunks; c += nthreads) {
        const int row = c / cpr;
        const int col = c - row * cpr;
        uint32_t ldst =
            (uint32_t)(size_t)&smem[lds_base_floats + row * lds_stride + col * 4];
        uint64_t src = (uint64_t)(size_t)(g + (size_t)row * row_len + col * 4);
        asm volatile("global_load_async_to_lds_b128 %0, %1, off"
                     :: "v"(ldst), "v"(src) : "memory");
    }
}

__device__ __forceinline__ void async_wait_and_sync() {
    asm volatile("s_wait_asynccnt 0x0" ::: "memory");
    __syncthreads();
}

// ---------------------------------------------------------------------------
// Kernel 1: gather k,v ; w = softmax(k@Mk^T) ; e = sigmoid(v@eW^T+eb) ;
//           a = tanh(v@aW^T+ab).  256 thr = 8 waves, 16 tokens per wave.
//           Weights staged once per block into LDS via async-to-LDS.
// ---------------------------------------------------------------------------
__global__ __launch_bounds__(256) void gea_kernel(
    const int* __restrict__ question, const int* __restrict__ respond,
    const float* __restrict__ k_emb, const float* __restrict__ v_emb,
    const float* __restrict__ Mk,
    const float* __restrict__ eW, const float* __restrict__ eb,
    const float* __restrict__ aW, const float* __restrict__ ab,
    float* __restrict__ w_out, float* __restrict__ e_out,
    float* __restrict__ a_out)
{
    const int tid  = threadIdx.x;
    const int lane = tid & 31;
    const int wave = tid >> 5;
    const int tok0 = (blockIdx.x * 8 + wave) * 16;
    const int lr   = lane & 15;   // row of A / col of B within tile
    const int hi   = lane >> 4;   // half-wave selector (K offset +2)

    // ---- stage Mk / eW / aW into LDS (async DMA path, ASYNCcnt) ----
    async_stage_rows(Mk, S_DIM, MK_BASE, LDS_STRIDE1, M_SIZE, tid, 256);
    async_stage_rows(eW, S_DIM, EW_BASE, LDS_STRIDE1, S_DIM, tid, 256);
    async_stage_rows(aW, S_DIM, AW_BASE, LDS_STRIDE1, S_DIM, tid, 256);

    const int q = question[tok0 + lr];
    const int x = q + NUM_Q * respond[tok0 + lr];

    // ---- A-operand: gathered k rows in WMMA f32 16x4 layout ----
    const float* rowK = k_emb + (size_t)q * S_DIM + 2 * hi;
    v2f Ak[32];
#pragma unroll
    for (int j = 0; j < 32; ++j)
        Ak[j] = *(const v2f*)(rowK + 4 * j);

    async_wait_and_sync();   // weights visible in LDS

    // ---- w logits: 4 n-tiles over M_SIZE=64, B from LDS ----
    v8f wacc[4];
#pragma unroll
    for (int nt = 0; nt < 4; ++nt) {
        const float* rowB = &smem[MK_BASE + (16 * nt + lr) * LDS_STRIDE1 + 2 * hi];
        v8f acc = {0.f, 0.f, 0.f, 0.f, 0.f, 0.f, 0.f, 0.f};
#pragma unroll
        for (int j = 0; j < 32; ++j)
            acc = wmma4(Ak[j], *(const v2f*)(rowB + 4 * j), acc);
        wacc[nt] = acc;
    }

    // ---- softmax over m (rows M and M+8 live in opposite 16-lane halves) ----
#pragma unroll
    for (int vg = 0; vg < 8; ++vg) {
        float mx = fmaxf(fmaxf(wacc[0][vg], wacc[1][vg]),
                         fmaxf(wacc[2][vg], wacc[3][vg]));
#pragma unroll
        for (int off = 1; off <= 8; off <<= 1)
            mx = fmaxf(mx, __shfl_xor(mx, off, 32));
        float sum = 0.f;
#pragma unroll
        for (int nt = 0; nt < 4; ++nt) {
            float ev = __expf(wacc[nt][vg] - mx);
            wacc[nt][vg] = ev;
            sum += ev;
        }
#pragma unroll
        for (int off = 1; off <= 8; off <<= 1)
            sum += __shfl_xor(sum, off, 32);
        float inv = 1.0f / sum;
#pragma unroll
        for (int nt = 0; nt < 4; ++nt)
            wacc[nt][vg] *= inv;
    }
#pragma unroll
    for (int nt = 0; nt < 4; ++nt)
#pragma unroll
        for (int vg = 0; vg < 8; ++vg)
            w_out[(size_t)(tok0 + vg + 8 * hi) * M_SIZE + 16 * nt + lr] =
                wacc[nt][vg];

    // ---- e / a GEMMs on gathered v rows, B from LDS ----
    const float* rowV = v_emb + (size_t)x * S_DIM + 2 * hi;
    v2f Av[32];
#pragma unroll
    for (int j = 0; j < 32; ++j)
        Av[j] = *(const v2f*)(rowV + 4 * j);

#pragma unroll
    for (int nt = 0; nt < 8; ++nt) {
        const float* rowE = &smem[EW_BASE + (16 * nt + lr) * LDS_STRIDE1 + 2 * hi];
        const float* rowA = &smem[AW_BASE + (16 * nt + lr) * LDS_STRIDE1 + 2 * hi];
        v8f eacc = {0.f, 0.f, 0.f, 0.f, 0.f, 0.f, 0.f, 0.f};
        v8f aacc = {0.f, 0.f, 0.f, 0.f, 0.f, 0.f, 0.f, 0.f};
#pragma unroll
        for (int j = 0; j < 32; ++j) {
            eacc = wmma4(Av[j], *(const v2f*)(rowE + 4 * j), eacc);
            aacc = wmma4(Av[j], *(const v2f*)(rowA + 4 * j), aacc);
        }
        const int   col = 16 * nt + lr;
        const float ebv = eb[col];
        const float abv = ab[col];
#pragma unroll
        for (int vg = 0; vg < 8; ++vg) {
            const size_t tkn = (size_t)(tok0 + vg + 8 * hi);
            e_out[tkn * S_DIM + col] = sigmoidf_(eacc[vg] + ebv);
            a_out[tkn * S_DIM + col] = tanhf(aacc[vg] + abv);
        }
    }
}

// ---------------------------------------------------------------------------
// Kernel 2: sequential DKVMN scan.  One block per batch, 512 thr = 16 waves.
// lane = (s%8)*4 + mc : lane owns Mv[mc*16 .. mc*16+15][s]; r-reduction over
// the 4 mc owners is two in-wave shfl_xor ops.  w_t double-buffered in LDS
// and prefetched one step ahead with async-to-LDS.
// ---------------------------------------------------------------------------
__global__ __launch_bounds__(512) void scan_kernel(
    const float* __restrict__ w_ws, const float* __restrict__ e_ws,
    const float* __restrict__ a_ws, const float* __restrict__ Mv0,
    float* __restrict__ r_ws)
{
    __shared__ float lw[2][M_SIZE];
    const int b   = blockIdx.x;
    const int tid = threadIdx.x;
    const int s   = tid >> 2;       // 0..127
    const int mc  = tid & 3;        // m-chunk 0..3

    float Mv[16];
#pragma unroll
    for (int i = 0; i < 16; ++i)
        Mv[i] = Mv0[(mc * 16 + i) * S_DIM + s];

    // prologue: stage w[t=0]
    if (tid < 16) {
        uint32_t ldst = (uint32_t)(size_t)&lw[0][tid * 4];
        uint64_t src  = (uint64_t)(size_t)(w_ws + (size_t)b * 512 * M_SIZE + tid * 4);
        asm volatile("global_load_async_to_lds_b128 %0, %1, off"
                     :: "v"(ldst), "v"(src) : "memory");
    }
    async_wait_and_sync();

    for (int t = 0; t < 512; ++t) {
        const int buf   = t & 1;
        const int token = b * 512 + t;

        if (t < 511 && tid < 16) {   // prefetch w[t+1] into the other buffer
            uint32_t ldst = (uint32_t)(size_t)&lw[buf ^ 1][tid * 4];
            uint64_t src  = (uint64_t)(size_t)
                (w_ws + (size_t)(token + 1) * M_SIZE + tid * 4);
            asm volatile("global_load_async_to_lds_b128 %0, %1, off"
                         :: "v"(ldst), "v"(src) : "memory");
        }

        const float ev = e_ws[(size_t)token * S_DIM + s];
        const float av = a_ws[(size_t)token * S_DIM + s];
        float r = 0.f;
#pragma unroll
        for (int i = 0; i < 16; ++i) {
            const float wm = lw[buf][mc * 16 + i];
            r = fmaf(wm, Mv[i], r);                       // read BEFORE write
            Mv[i] = fmaf(wm, av, fmaf(-wm * ev, Mv[i], Mv[i]));
        }
        r += __shfl_xor(r, 1, 32);
        r += __shfl_xor(r, 2, 32);
        if (mc == 0)
            r_ws[(size_t)token * S_DIM + s] = r;

        asm volatile("s_wait_asynccnt 0x0" ::: "memory");
        __syncthreads();
    }
}

// ---------------------------------------------------------------------------
// Kernel 3: f = tanh([r,k] @ fW^T + fb) ; p = sigmoid(f @ pW^T + pb).
// 256 thr = 8 waves, 16 tokens per wave; fW staged into LDS.
// ---------------------------------------------------------------------------
__global__ __launch_bounds__(256) void final_kernel(
    const int* __restrict__ question, const float* __restrict__ k_emb,
    const float* __restrict__ r_ws,
    const float* __restrict__ fW, const float* __restrict__ fb,
    const float* __restrict__ pW, const float* __restrict__ pb,
    float* __restrict__ p_out)
{
    const int tid  = threadIdx.x;
    const int lane = tid & 31;
    const int wave = tid >> 5;
    const int tok0 = (blockIdx.x * 8 + wave) * 16;
    const int lr   = lane & 15;
    const int hi   = lane >> 4;

    async_stage_rows(fW, 2 * S_DIM, 0, LDS_STRIDE3, S_DIM, tid, 256);

    const int    q    = question[tok0 + lr];
    const float* rowR = r_ws + (size_t)(tok0 + lr) * S_DIM + 2 * hi;
    const float* rowK = k_emb + (size_t)q * S_DIM + 2 * hi;

    v2f Ar[32], Ak[32];
#pragma unroll
    for (int j = 0; j < 32; ++j) {
        Ar[j] = *(const v2f*)(rowR + 4 * j);
        Ak[j] = *(const v2f*)(rowK + 4 * j);
    }

    async_wait_and_sync();

    float pdot[8] = {0.f, 0.f, 0.f, 0.f, 0.f, 0.f, 0.f, 0.f};

#pragma unroll
    for (int nt = 0; nt < 8; ++nt) {
        const float* rowF = &smem[(16 * nt + lr) * LDS_STRIDE3 + 2 * hi];
        v8f acc = {0.f, 0.f, 0.f, 0.f, 0.f, 0.f, 0.f, 0.f};
#pragma unroll
        for (int j = 0; j < 32; ++j)                      // r-half of concat
            acc = wmma4(Ar[j], *(const v2f*)(rowF + 4 * j), acc);
#pragma unroll
        for (int j = 0; j < 32; ++j)                      // k-half of concat
            acc = wmma4(Ak[j], *(const v2f*)(rowF + S_DIM + 4 * j), acc);

        const int   col = 16 * nt + lr;
        const float fbv = fb[col];
        const float pwv = pW[col];
#pragma unroll
        for (int vg = 0; vg < 8; ++vg) {
            const float fv = tanhf(acc[vg] + fbv);
            pdot[vg] = fmaf(fv, pwv, pdot[vg]);
        }
    }

    const float pbv = pb[0];
#pragma unroll
    for (int vg = 0; vg < 8; ++vg) {
#pragma unroll
        for (int off = 1; off <= 8; off <<= 1)
            pdot[vg] += __shfl_xor(pdot[vg], off, 32);
        if (lr == vg)
            p_out[tok0 + vg + 8 * hi] = sigmoidf_(pdot[vg] + pbv);
    }
}

// ---------------------------------------------------------------------------
extern "C" void kernel_launch(void* const* d_in, const int* in_sizes, int n_in,
                              void* d_out, int out_size, void* d_ws,
                              size_t ws_size, hipStream_t stream)
{
    const int*   question = (const int*)d_in[0];
    const int*   respond  = (const int*)d_in[1];
    const float* k_emb    = (const float*)d_in[2];
    const float* v_emb    = (const float*)d_in[3];
    const float* Mk       = (const float*)d_in[4];
    const float* Mv0      = (const float*)d_in[5];
    const float* eW       = (const float*)d_in[6];
    const float* eb       = (const float*)d_in[7];
    const float* aW       = (const float*)d_in[8];
    const float* ab       = (const float*)d_in[9];
    const float* fW       = (const float*)d_in[10];
    const float* fb       = (const float*)d_in[11];
    const float* pW       = (const float*)d_in[12];
    const float* pb       = (const float*)d_in[13];

    float* ws   = (float*)d_ws;
    float* w_ws = ws;                                  // [NTOK, 64]   8 MB
    float* e_ws = w_ws + (size_t)NTOK * M_SIZE;        // [NTOK, 128] 16 MB
    float* a_ws = e_ws + (size_t)NTOK * S_DIM;         // [NTOK, 128] 16 MB
    float* r_ws = a_ws + (size_t)NTOK * S_DIM;         // [NTOK, 128] 16 MB

    const int lds1 = (64 + 128 + 128) * LDS_STRIDE1 * 4;   // 168960 B
    const int lds3 = 128 * LDS_STRIDE3 * 4;                // 133120 B
    hipFuncSetAttribute((const void*)gea_kernel,
                        hipFuncAttributeMaxDynamicSharedMemorySize, lds1);
    hipFuncSetAttribute((const void*)final_kernel,
                        hipFuncAttributeMaxDynamicSharedMemorySize, lds3);

    // 32768 tokens / 16 per wave / 8 waves per block = 256 blocks
    gea_kernel<<<256, 256, lds1, stream>>>(question, respond, k_emb, v_emb, Mk,
                                           eW, eb, aW, ab, w_ws, e_ws, a_ws);
    scan_kernel<<<64, 512, 0, stream>>>(w_ws, e_ws, a_ws, Mv0, r_ws);
    final_kernel<<<256, 256, lds3, stream>>>(question, k_emb, r_ws, fW, fb, pW,
                                             pb, (float*)d_out);
}